// BackboneDistanceEmbedding_43439299232114
// MI455X (gfx1250) — compile-verified
//
#include <hip/hip_runtime.h>
#include <math.h>

#define N_NODES 8192
#define KZ 20
#define PD 20
#define NVZ 64
#define NUM_PROT 20
#define FZ 256
#define ENC_LEN 2000   /* 5*KZ*PD */
#define NFEAT 100      /* 5*KZ   */

typedef __attribute__((ext_vector_type(2))) float v2f;
typedef __attribute__((ext_vector_type(8))) float v8f;

// freq[m] = 50^(-m/10) = exp(-ln(50)/10 * m),  m = 0..9
__device__ __forceinline__ float freq_of(int m) {
    return __expf(-0.39120230054281461f * (float)m);
}

// ---------------------------------------------------------------------------
// Kernel 1: per-node setup.  positions out, (x,y,z,|p|^2) staging, ncac.
// ncac[a][i] = sum_j rot[i][j]*lit[aatype][a][j] + trans[i]
// ---------------------------------------------------------------------------
__global__ __launch_bounds__(256) void k_setup(
    const float* __restrict__ affines,        // N x 3 x 4
    const float* __restrict__ lit_positions,  // 21 x 3 x 3
    const int*   __restrict__ prot_mask,      // N (bool as int)
    float*  __restrict__ out_positions,       // N x 3
    float4* __restrict__ ws_pos4,             // N
    float*  __restrict__ ws_ncac)             // N x 9
{
    int n = blockIdx.x * blockDim.x + threadIdx.x;
    if (n >= N_NODES) return;
    const float* A = affines + (size_t)n * 12;
    float r00=A[0], r01=A[1], r02=A[2],  tx=A[3];
    float r10=A[4], r11=A[5], r12=A[6],  ty=A[7];
    float r20=A[8], r21=A[9], r22=A[10], tz=A[11];
    out_positions[n*3+0]=tx; out_positions[n*3+1]=ty; out_positions[n*3+2]=tz;
    ws_pos4[n] = make_float4(tx, ty, tz, tx*tx + ty*ty + tz*tz);
    int aat = (prot_mask[n] != 0) ? 0 : NUM_PROT;
    const float* L = lit_positions + aat * 9;
    #pragma unroll
    for (int a = 0; a < 3; ++a) {
        float lx = L[a*3+0], ly = L[a*3+1], lz = L[a*3+2];
        ws_ncac[(size_t)n*9 + a*3 + 0] = r00*lx + r01*ly + r02*lz + tx;
        ws_ncac[(size_t)n*9 + a*3 + 1] = r10*lx + r11*ly + r12*lz + ty;
        ws_ncac[(size_t)n*9 + a*3 + 2] = r20*lx + r21*ly + r22*lz + tz;
    }
}

// ---------------------------------------------------------------------------
// Kernel 2: exact KNN, one thread per query row.  Candidate tiles are staged
// into LDS with the CDNA5 async memory->LDS path (no VGPR round trip):
// global_load_async_to_lds_b128 + s_wait_asynccnt.
// score = |q|^2 - 2 p.q  (row-constant |p|^2 dropped: top-k invariant).
// Sorted ascending 20-entry register list; strict < keeps lower index on tie.
// ---------------------------------------------------------------------------
__global__ __launch_bounds__(256) void k_knn(
    const float4* __restrict__ pos4,
    int*   __restrict__ ws_edge,         // N x KZ
    float* __restrict__ out_edge,        // N x KZ (float-converted)
    float* __restrict__ out_full_edge)   // 2 x (N*KZ)
{
    __shared__ float4 sp[256];
    int r = blockIdx.x * 256 + threadIdx.x;
    float4 p = pos4[r];
    float bd[KZ]; int bi[KZ];
    #pragma unroll
    for (int t = 0; t < KZ; ++t) { bd[t] = 3.0e38f; bi[t] = 0; }

    // generic shared pointer: low 32 bits are the LDS byte offset
    unsigned lds_off = (unsigned)(size_t)(&sp[threadIdx.x]);

    for (int tile = 0; tile < N_NODES; tile += 256) {
        __syncthreads();   // previous tile fully consumed
        unsigned long long ga = (unsigned long long)(pos4 + tile + threadIdx.x);
        asm volatile("global_load_async_to_lds_b128 %0, %1, off"
                     :: "v"(lds_off), "v"(ga) : "memory");
        asm volatile("s_wait_asynccnt 0" ::: "memory");
        __syncthreads();   // whole tile resident in LDS
        for (int jj = 0; jj < 256; ++jj) {
            float4 q = sp[jj];
            int j = tile + jj;
            float s = q.w - 2.0f * (p.x*q.x + p.y*q.y + p.z*q.z);
            if (s < bd[KZ-1] && j != r) {
                bd[KZ-1] = s; bi[KZ-1] = j;
                #pragma unroll
                for (int t = KZ-1; t > 0; --t) {
                    if (bd[t] < bd[t-1]) {
                        float td = bd[t]; bd[t] = bd[t-1]; bd[t-1] = td;
                        int   ti = bi[t]; bi[t] = bi[t-1]; bi[t-1] = ti;
                    }
                }
            }
        }
    }
    #pragma unroll
    for (int t = 0; t < KZ; ++t) {
        ws_edge[(size_t)r*KZ + t]  = bi[t];
        out_edge[(size_t)r*KZ + t] = (float)bi[t];
        out_full_edge[(size_t)r*KZ + t] = (float)bi[t];
        out_full_edge[(size_t)N_NODES*KZ + (size_t)r*KZ + t] = (float)r;
    }
}

// ---------------------------------------------------------------------------
// Kernel 3: neighbour_positions (rot^T applied) + sinusoidal distance enc.
// One thread per (n,k).
// ---------------------------------------------------------------------------
__global__ __launch_bounds__(256) void k_nbr(
    const float*  __restrict__ affines,
    const float4* __restrict__ pos4,
    const int*    __restrict__ ws_edge,
    float* __restrict__ out_np,   // N x KZ x 3
    float* __restrict__ out_nd)   // N x KZ x PD
{
    int g = blockIdx.x * blockDim.x + threadIdx.x;
    if (g >= N_NODES * KZ) return;
    int n = g / KZ;
    const float* A = affines + (size_t)n * 12;
    float tx = A[3], ty = A[7], tz = A[11];
    int j = ws_edge[g];
    float4 q = pos4[j];
    float b0 = q.x - tx, b1 = q.y - ty, b2 = q.z - tz;
    // einsum 'nji,nkj->nki': out_i = sum_j rot[j][i] * b_j
    float o0 = A[0]*b0 + A[4]*b1 + A[8]*b2;
    float o1 = A[1]*b0 + A[5]*b1 + A[9]*b2;
    float o2 = A[2]*b0 + A[6]*b1 + A[10]*b2;
    out_np[(size_t)g*3 + 0] = o0;
    out_np[(size_t)g*3 + 1] = o1;
    out_np[(size_t)g*3 + 2] = o2;
    float d = sqrtf(o0*o0 + o1*o1 + o2*o2);
    #pragma unroll
    for (int m = 0; m < 10; ++m) {
        float ang = d * freq_of(m);
        out_nd[(size_t)g*PD + m]      = __sinf(ang);
        out_nd[(size_t)g*PD + 10 + m] = __cosf(ang);
    }
}

// ---------------------------------------------------------------------------
// Kernel 4: fused feature build + sinusoidal enc + GEMM via
// V_WMMA_F32_16X16X4_F32.  128 threads (4 waves) per block, 16 nodes.
// Phase 1: 16x100 distance features -> LDS.
// Phase 2: full 16x2000 enc tile -> dynamic LDS (128 KB; 320 KB WGP LDS
//          makes this cheap), each enc value computed exactly once with
//          uniform (non-divergent) v_sin/v_cos.
// Phase 3: per-wave 16x16 output tile, K swept 4 at a time:
//          ds_load_b64 (A) + global_load_b64 (B, L2-resident weights) + wmma.
// Also copies x into x_ne[:, :256].
// ---------------------------------------------------------------------------
__global__ __launch_bounds__(128) void k_embed(
    const float* __restrict__ x,          // N x 256
    const float* __restrict__ ne_weight,  // 64 x 2000
    const int*   __restrict__ ws_edge,    // N x KZ
    const float* __restrict__ ws_ncac,    // N x 9
    float* __restrict__ out_xne)          // N x 320
{
    extern __shared__ float smem_f[];
    float* enc_lds = smem_f;                    // 16 * 2000
    float* f_lds   = smem_f + 16 * ENC_LEN;     // 16 * 100
    __shared__ float freq_s[10];

    int nb  = blockIdx.x * 16;
    int tid = threadIdx.x;

    if (tid < 10) freq_s[tid] = freq_of(tid);

    // copy x rows -> x_ne columns [0,256), float4 moves
    const float4* x4 = (const float4*)x;
    float4*       o4 = (float4*)out_xne;
    for (int e = tid; e < 16 * 64; e += 128) {
        int row = e >> 6, c = e & 63;
        o4[(size_t)(nb + row) * 80 + c] = x4[(size_t)(nb + row) * 64 + c];
    }

    // Phase 1: features [ca_to_ncac(60) | n_to_c(20) | c_to_n(20)] per node
    for (int t = tid; t < 16 * KZ; t += 128) {
        int m = t / KZ, k = t % KZ;
        int n = nb + m;
        int j = ws_edge[(size_t)n * KZ + k];
        const float* nc = ws_ncac + (size_t)j * 9;
        const float* oc = ws_ncac + (size_t)n * 9;
        float npx = oc[0], npy = oc[1], npz = oc[2];   // N atom
        float cax = oc[3], cay = oc[4], caz = oc[5];   // CA atom
        float cpx = oc[6], cpy = oc[7], cpz = oc[8];   // C atom
        #pragma unroll
        for (int a = 0; a < 3; ++a) {
            float dx = nc[a*3+0]-cax, dy = nc[a*3+1]-cay, dz = nc[a*3+2]-caz;
            f_lds[m*NFEAT + k*3 + a] = sqrtf(dx*dx + dy*dy + dz*dz);
        }
        { float dx = nc[6]-npx, dy = nc[7]-npy, dz = nc[8]-npz;       // nbr C - own N
          f_lds[m*NFEAT + 60 + k] = sqrtf(dx*dx + dy*dy + dz*dz); }
        { float dx = nc[0]-cpx, dy = nc[1]-cpy, dz = nc[2]-cpz;       // nbr N - own C
          f_lds[m*NFEAT + 80 + k] = sqrtf(dx*dx + dy*dy + dz*dz); }
    }
    __syncthreads();

    // Phase 2: enc tile.  Task = (node m, feature i); 10 uniform sin+cos each.
    for (int t = tid; t < 16 * NFEAT; t += 128) {
        int m = t / NFEAT, i = t - m * NFEAT;
        float fv = f_lds[t];                       // layout matches m*NFEAT+i
        float* dst = enc_lds + m * ENC_LEN + i * PD;
        #pragma unroll
        for (int q = 0; q < 10; ++q) {
            float ang = fv * freq_s[q];
            dst[q]      = __sinf(ang);
            dst[10 + q] = __cosf(ang);
        }
    }
    __syncthreads();

    // Phase 3: WMMA.  A 16x4 f32: lanes 0-15 hold K0,K1; lanes 16-31 K2,K3.
    int lane  = tid & 31;
    int wave  = tid >> 5;            // wave32: 4 waves -> 4 col tiles
    int m     = lane & 15;           // A row / B,D column within tile
    int khalf = (lane >> 4) * 2;     // 0 or 2
    int v     = wave * 16 + m;       // output column 0..63

    v8f acc = {0.f, 0.f, 0.f, 0.f, 0.f, 0.f, 0.f, 0.f};
    const float* em   = enc_lds + m * ENC_LEN;
    const float* wrow = ne_weight + (size_t)v * ENC_LEN;

    #pragma unroll 4
    for (int K0 = 0; K0 < ENC_LEN; K0 += 4) {
        int ka = K0 + khalf;                        // even -> 8B aligned
        v2f a = *(const v2f*)(em + ka);             // ds_load_b64
        float2 wv = *(const float2*)(wrow + ka);    // global_load_b64
        v2f b; b[0] = wv.x; b[1] = wv.y;
        acc = __builtin_amdgcn_wmma_f32_16x16x4_f32(
                  false, a, false, b, (short)0, acc, false, false);
    }

    int rowhi = (lane >> 4) * 8;     // D: lanes 16-31 hold M=8..15
    #pragma unroll
    for (int r = 0; r < 8; ++r) {
        int node = nb + r + rowhi;
        out_xne[(size_t)node * 320 + 256 + v] = acc[r];
    }
}

// ---------------------------------------------------------------------------
extern "C" void kernel_launch(void* const* d_in, const int* in_sizes, int n_in,
                              void* d_out, int out_size, void* d_ws, size_t ws_size,
                              hipStream_t stream) {
    const float* x             = (const float*)d_in[0];
    const float* affines       = (const float*)d_in[1];
    const float* ne_weight     = (const float*)d_in[2];
    const float* lit_positions = (const float*)d_in[3];
    const int*   prot_mask     = (const int*)d_in[4];

    float* out     = (float*)d_out;
    float* out_xne = out;                 // 8192*320 = 2,621,440
    float* out_pos = out + 2621440;       // 8192*3
    float* out_np  = out + 2646016;       // 8192*20*3
    float* out_nd  = out + 3137536;       // 8192*20*20
    float* out_ei  = out + 6414336;       // 8192*20
    float* out_fei = out + 6578176;       // 2*8192*20

    char*   ws      = (char*)d_ws;
    float4* ws_pos4 = (float4*)ws;                                  // 128 KiB
    float*  ws_ncac = (float*)(ws + (size_t)N_NODES * 16);          // 288 KiB
    int*    ws_edge = (int*)  (ws + (size_t)N_NODES * 16
                                  + (size_t)N_NODES * 36);          // 640 KiB

    k_setup<<<N_NODES / 256, 256, 0, stream>>>(affines, lit_positions, prot_mask,
                                               out_pos, ws_pos4, ws_ncac);
    k_knn<<<N_NODES / 256, 256, 0, stream>>>(ws_pos4, ws_edge, out_ei, out_fei);
    k_nbr<<<(N_NODES * KZ + 255) / 256, 256, 0, stream>>>(affines, ws_pos4, ws_edge,
                                                          out_np, out_nd);
    size_t smem = (size_t)(16 * ENC_LEN + 16 * NFEAT) * sizeof(float);  // 134.4 KB
    k_embed<<<N_NODES / 16, 128, smem, stream>>>(x, ne_weight, ws_edge, ws_ncac,
                                                 out_xne);
}